// NicheAttention_51994874085932
// MI455X (gfx1250) — compile-verified
//
#include <hip/hip_runtime.h>

typedef float v2f __attribute__((ext_vector_type(2)));
typedef float v8f __attribute__((ext_vector_type(8)));

// ---------------------------------------------------------------- init ------
__global__ void init_kernel(float* __restrict__ out, float* __restrict__ segmax,
                            float* __restrict__ denom, int BD, int B) {
  int i = blockIdx.x * blockDim.x + threadIdx.x;
  if (i < BD) out[i] = 0.0f;
  if (i < B) {
    segmax[i] = __int_as_float(0xFF800000); // -inf
    denom[i]  = 0.0f;
  }
}

// ------------------------------------------------ scores + segment max ------
__global__ __launch_bounds__(256)
void scores_kernel(const float* __restrict__ x, const float* __restrict__ w,
                   const float* __restrict__ bias, const int* __restrict__ batch,
                   float* __restrict__ scores, float* __restrict__ segmax,
                   int N) {
  const int lane = threadIdx.x & 31;
  const int gw   = blockIdx.x * (blockDim.x >> 5) + (threadIdx.x >> 5);
  const int nw   = gridDim.x * (blockDim.x >> 5);
  const float4* w4 = (const float4*)w;          // D == 256 -> 64 float4
  const float4 wa = w4[lane];
  const float4 wb = w4[lane + 32];
  const float  b  = bias[0];
  for (int row = gw; row < N; row += nw) {
    const float4* xr = (const float4*)(x + (size_t)row * 256);
    float4 xa = xr[lane];
    float4 xb = xr[lane + 32];
    float s = xa.x * wa.x + xa.y * wa.y + xa.z * wa.z + xa.w * wa.w
            + xb.x * wb.x + xb.y * wb.y + xb.z * wb.z + xb.w * wb.w;
    #pragma unroll
    for (int off = 16; off > 0; off >>= 1) s += __shfl_xor(s, off, 32);
    if (lane == 0) {
      s += b;
      scores[row] = s;
      float* addr = segmax + batch[row];
      if (s >= 0.0f) atomicMax((int*)addr, __float_as_int(s));
      else           atomicMin((unsigned int*)addr, __float_as_uint(s));
    }
  }
}

// ------------------------------------------------ exp + segment sum ---------
__global__ void softmax_kernel(const int* __restrict__ batch,
                               const float* __restrict__ segmax,
                               float* __restrict__ scores,  // in: score, out: e
                               float* __restrict__ denom, int N) {
  int i = blockIdx.x * blockDim.x + threadIdx.x;
  if (i >= N) return;
  int s = batch[i];
  float e = expf(scores[i] - segmax[s]);
  scores[i] = e;
  atomicAdd(&denom[s], e);
}

// ---------------------------------------------- rdenom = 1/denom ------------
__global__ void invert_kernel(float* __restrict__ denom, int B) {
  int i = blockIdx.x * blockDim.x + threadIdx.x;
  if (i < B) denom[i] = 1.0f / denom[i];
}

// ---------------------------------------------- attn = e * rdenom[seg] ------
__global__ void attn_kernel(const int* __restrict__ batch,
                            const float* __restrict__ rdenom,
                            float* __restrict__ e_attn, int N) {
  int i = blockIdx.x * blockDim.x + threadIdx.x;
  if (i < N) e_attn[i] = e_attn[i] * rdenom[batch[i]];
}

// ------------------------- WMMA scatter-weighted segment sum ----------------
// Rows [0, N32), N32 % 32 == 0 -> no range guards in the streaming loop.
// Block = 256 threads = 8 waves; wave w owns columns [32w, 32w+32).
// Row<->K mapping: row(K,s) = r0 + 8K + s, so A operands are DPP ROW_SHARE
// broadcasts (no LDS) and B rows are constant-offset b64 loads per lane.
// V_WMMA_F32_16X16X4_F32, two accumulators: even cols / odd cols.
// A[m][k] = attn(row) if batch(row) == tb+m else 0 (one-hot scatter matmul).

// broadcast lane (0..15) within each row of 16 lanes: v_mov_b32_dpp row_share
#define DPP_ROWSHARE(v, l) \
  __builtin_amdgcn_update_dpp(0, (v), 0x150 + (l), 0xF, 0xF, false)

__global__ __launch_bounds__(256)
void wsum_kernel(const float* __restrict__ x, const int* __restrict__ batch,
                 const float* __restrict__ attn, float* __restrict__ out,
                 int N32, int B, int rowsPerBlock) {
  const int D    = 256;
  const int lane = threadIdx.x & 31;
  const int wcol = (threadIdx.x >> 5) * 32;   // wave's 32 columns
  const int n    = lane & 15;                 // A slot m / col-pair index
  const int hi   = lane >> 4;                 // 0: lanes 0-15, 1: lanes 16-31

  const int rowBeg = blockIdx.x * rowsPerBlock;
  const int rowEnd = min(rowBeg + rowsPerBlock, N32);
  const float* __restrict__ xcol = x + wcol + 2 * n;  // this lane's column pair

  v8f c0, c1;
  #pragma unroll
  for (int i = 0; i < 8; ++i) { c0[i] = 0.0f; c1[i] = 0.0f; }
  int tb = -1;  // segment base of current accumulator window

  for (int r0 = rowBeg; r0 < rowEnd; r0 += 32) {
    const int myRow   = r0 + lane;
    const int   myseg  = batch[myRow];
    const float myattn = attn[myRow];
    const int   attn_i = __float_as_int(myattn);

    // batch sorted -> chunk min = lane 0, chunk max = lane 31 (both uniform)
    const int cmin = __builtin_amdgcn_readfirstlane(myseg);
    const int cmax = __builtin_amdgcn_readlane(myseg, 31);

    if (tb < 0) tb = cmin;
    if (cmax >= tb + 16) {
      // flush accumulators into out with f32 atomics, restart window
      #pragma unroll
      for (int r = 0; r < 8; ++r) {
        const int seg = tb + r + hi * 8;
        if (seg < B) {
          if (c0[r] != 0.0f) atomicAdd(&out[(size_t)seg * D + wcol + 2 * n],     c0[r]);
          if (c1[r] != 0.0f) atomicAdd(&out[(size_t)seg * D + wcol + 2 * n + 1], c1[r]);
        }
        c0[r] = 0.0f; c1[r] = 0.0f;
      }
      tb = cmin;
    }

    // rare: >16 distinct segments inside one 32-row chunk -> direct atomics.
    // (such rows can never match tb+m in the A build, so no masking needed)
    const bool fb = (myseg >= tb + 16);
    if (__any((int)fb)) {
      if (fb) {
        const float* xr = x + (size_t)myRow * D + wcol;
        float* orow = out + (size_t)myseg * D + wcol;
        for (int j = 0; j < 32; ++j) atomicAdd(&orow[j], myattn * xr[j]);
      }
    }

    const int tbn = tb + n;
    const float* __restrict__ xrow = xcol + (size_t)(r0 + 16 * hi) * D;

    // 8 WMMA K-steps; this lane's K rows: r0+16hi+s (K=2hi), r0+16hi+8+s (K=2hi+1)
#define KSTEP(s)                                                               \
    {                                                                          \
      const int a0s = DPP_ROWSHARE(myseg, (s));                                \
      const int a1s = DPP_ROWSHARE(myseg, (s) + 8);                            \
      const int a0i = DPP_ROWSHARE(attn_i, (s));                               \
      const int a1i = DPP_ROWSHARE(attn_i, (s) + 8);                           \
      v2f A;                                                                   \
      A.x = (a0s == tbn) ? __int_as_float(a0i) : 0.0f;                         \
      A.y = (a1s == tbn) ? __int_as_float(a1i) : 0.0f;                         \
      const float2 pa = *(const float2*)(xrow + (s) * 256);                    \
      const float2 pb = *(const float2*)(xrow + ((s) + 8) * 256);              \
      v2f Be, Bo;                                                              \
      Be.x = pa.x; Be.y = pb.x;  /* even cols */                               \
      Bo.x = pa.y; Bo.y = pb.y;  /* odd cols  */                               \
      c0 = __builtin_amdgcn_wmma_f32_16x16x4_f32(false, A, false, Be,          \
                                                 (short)0, c0, false, false);  \
      c1 = __builtin_amdgcn_wmma_f32_16x16x4_f32(false, A, false, Bo,          \
                                                 (short)0, c1, false, false);  \
    }
    KSTEP(0) KSTEP(1) KSTEP(2) KSTEP(3) KSTEP(4) KSTEP(5) KSTEP(6) KSTEP(7)
#undef KSTEP
  }

  // final flush
  if (tb >= 0) {
    #pragma unroll
    for (int r = 0; r < 8; ++r) {
      const int seg = tb + r + hi * 8;
      if (seg < B) {
        if (c0[r] != 0.0f) atomicAdd(&out[(size_t)seg * D + wcol + 2 * n],     c0[r]);
        if (c1[r] != 0.0f) atomicAdd(&out[(size_t)seg * D + wcol + 2 * n + 1], c1[r]);
      }
    }
  }
}

// ----------------------- residual rows (N % 32 != 0 only) -------------------
__global__ void tail_kernel(const float* __restrict__ x, const int* __restrict__ batch,
                            const float* __restrict__ attn, float* __restrict__ out,
                            int N32, int N) {
  const int D = 256;
  int i = blockIdx.x * blockDim.x + threadIdx.x;
  int total = (N - N32) * D;
  if (i < total) {
    int row = N32 + i / D;
    int col = i % D;
    atomicAdd(&out[(size_t)batch[row] * D + col], attn[row] * x[(size_t)row * D + col]);
  }
}

// ---------------------------------------------------------------------------
extern "C" void kernel_launch(void* const* d_in, const int* in_sizes, int n_in,
                              void* d_out, int out_size, void* d_ws, size_t ws_size,
                              hipStream_t stream) {
  (void)n_in; (void)ws_size;
  const float* x     = (const float*)d_in[0];
  const float* w     = (const float*)d_in[1];
  const float* bias  = (const float*)d_in[2];
  const int*   batch = (const int*)d_in[3];

  const int N  = in_sizes[3];
  const int D  = in_sizes[0] / N;     // 256
  const int B  = out_size / D;        // 4096
  const int BD = B * D;
  const int N32 = N & ~31;            // full 32-row chunks

  float* out    = (float*)d_out;
  float* ws_e   = (float*)d_ws;       // N floats: scores -> exp -> attn
  float* segmax = ws_e + N;           // B floats
  float* denom  = segmax + B;         // B floats (later: 1/denom)

  // 0) init out / segmax / denom
  init_kernel<<<(BD + 255) / 256, 256, 0, stream>>>(out, segmax, denom, BD, B);

  // 1) scores = x@w + b, atomic segment max
  scores_kernel<<<4096, 256, 0, stream>>>(x, w, bias, batch, ws_e, segmax, N);

  // 2) e = exp(score - segmax[seg]), denom[seg] += e
  softmax_kernel<<<(N + 255) / 256, 256, 0, stream>>>(batch, segmax, ws_e, denom, N);

  // 2a) denom -> 1/denom
  invert_kernel<<<(B + 255) / 256, 256, 0, stream>>>(denom, B);

  // 2b) attn = e * rdenom[seg]   (in place in ws_e)
  attn_kernel<<<(N + 255) / 256, 256, 0, stream>>>(batch, denom, ws_e, N);

  // 3) out[seg] += attn * x  via f32 WMMA one-hot matmul (rows [0, N32))
  const int rowsPerBlock = 512;       // multiple of 32
  const int nb3 = (N32 + rowsPerBlock - 1) / rowsPerBlock;
  if (nb3 > 0)
    wsum_kernel<<<nb3, 256, 0, stream>>>(x, batch, ws_e, out, N32, B, rowsPerBlock);

  // 3b) residual rows (none when N % 32 == 0)
  if (N32 < N) {
    const int tailElems = (N - N32) * D;
    tail_kernel<<<(tailElems + 255) / 256, 256, 0, stream>>>(x, batch, ws_e, out, N32, N);
  }
}